// Postprocess_31301721653676
// MI455X (gfx1250) — compile-verified
//
#include <hip/hip_runtime.h>
#include <stdint.h>

#define MH 256
#define MW 256
#define MHW 65536            // pixels per mask
#define LSENT 65536          // background sentinel label (== cv2 stats row for bg)
#define NTH 1024             // 32 wave32 waves per block
#define PPT 64               // pixels per thread
#define CH 16352             // histogram bins per pass (5 passes cover 65536 labels)
#define NPASS 5

#define LBL_BYTES (MHW * 4)          // 262144
#define UNI_BYTES (CH * 4)           // 65408 (staging / histogram / reduce scratch)
#define MISC_BYTES 128
#define SMEM_BYTES (LBL_BYTES + UNI_BYTES + MISC_BYTES)  // 327680 = 320 KB (max LDS/WG)

// ---------------- top-2 (area desc, label asc) — matches lax.top_k tie-break ----
struct Top2 { int a1, l1, a2, l2; };

__device__ __forceinline__ bool t2_better(int a, int l, int b, int m) {
  return (a > b) || (a == b && l < m);
}
__device__ __forceinline__ void t2_push(Top2& c, int a, int l) {
  if (l == c.l1 || l == c.l2) return;           // dedupe (labels are unique keys)
  if (t2_better(a, l, c.a1, c.l1)) { c.a2 = c.a1; c.l2 = c.l1; c.a1 = a; c.l1 = l; }
  else if (t2_better(a, l, c.a2, c.l2)) { c.a2 = a; c.l2 = l; }
}

// ---------------- CDNA5 async global->LDS DMA (ASYNCcnt path) -------------------
__device__ __forceinline__ void async_wait0() {
#if defined(__AMDGCN__)
#if __has_builtin(__builtin_amdgcn_s_wait_asynccnt)
  __builtin_amdgcn_s_wait_asynccnt(0);
  asm volatile("" ::: "memory");
#else
  asm volatile("s_wait_asynccnt 0" ::: "memory");
#endif
#endif
}

__device__ __forceinline__ void async_g2l_f32(const float* g, float* l) {
#if defined(__AMDGCN__) && __has_builtin(__builtin_amdgcn_global_load_async_to_lds_b32)
  __builtin_amdgcn_global_load_async_to_lds_b32(
      (__attribute__((address_space(1))) int*)(uintptr_t)g,
      (__attribute__((address_space(3))) int*)(uint32_t)(uintptr_t)l,
      0, 0);
#elif defined(__AMDGCN__)
  unsigned lofs = (unsigned)(uintptr_t)l;       // generic LDS addr low 32b == LDS offset
  asm volatile("global_load_async_to_lds_b32 %0, %1, off"
               :: "v"(lofs), "v"(g) : "memory");
#else
  *l = *g;
#endif
}

// One block per mask: whole 256x256 label field lives in LDS (256 KB of 320 KB/WGP).
__global__ __launch_bounds__(NTH, 1)
void ccl_second_largest_kernel(const float* __restrict__ masks,
                               float* __restrict__ out) {
  extern __shared__ char smem[];
  int* lbl  = (int*)smem;                          // [65536] labels
  int* uni  = (int*)(smem + LBL_BYTES);            // staging / histogram / scratch
  int* misc = (int*)(smem + LBL_BYTES + UNI_BYTES);// [0]=changed [1]=fgTotal [2]=lab2 [3..6]=bbox

  const int t    = threadIdx.x;
  const int wid  = t >> 5;
  const int lane = t & 31;
  const float* src = masks + (size_t)blockIdx.x * MHW;
  float*       dst = out   + (size_t)blockIdx.x * MHW;

  // ---- Phase 1: async-DMA mask into LDS staging, init labels (fg ? i : SENT) ----
  float* stage = (float*)uni;                      // 8192-float chunks (32 KB)
  for (int q = 0; q < 8; ++q) {
#pragma unroll
    for (int s = 0; s < 8; ++s) {
      int i = t + (((q << 3) + s) << 10);
      async_g2l_f32(src + i, stage + (s << 10) + t);
    }
    async_wait0();
#pragma unroll
    for (int s = 0; s < 8; ++s) {
      int i = t + (((q << 3) + s) << 10);
      lbl[i] = (stage[(s << 10) + t] > 0.0f) ? i : LSENT;
    }
    __syncthreads();
  }

  // ---- Phase 2: min-label propagation + pointer jumping to fixed point ----------
  // In-place chaotic updates are monotone-decreasing and converge to the same
  // fixed point as the reference (min pixel index per 8-connected component).
  // A barrier-synced sweep with zero changes certifies the fixed point exactly.
  for (;;) {
    if (t == 0) misc[0] = 0;
    __syncthreads();
    int changed = 0;
#pragma unroll 2
    for (int k = 0; k < PPT; ++k) {              // 8-neighborhood min sweep
      int i = t + (k << 10);
      int v = lbl[i];
      if (v < LSENT) {
        int r = i >> 8, c = i & 255;
        int m = v;
        if (r > 0) {
          m = min(m, lbl[i - 256]);
          if (c > 0)   m = min(m, lbl[i - 257]);
          if (c < 255) m = min(m, lbl[i - 255]);
        }
        if (c > 0)   m = min(m, lbl[i - 1]);
        if (c < 255) m = min(m, lbl[i + 1]);
        if (r < 255) {
          m = min(m, lbl[i + 256]);
          if (c > 0)   m = min(m, lbl[i + 255]);
          if (c < 255) m = min(m, lbl[i + 257]);
        }
        if (m < v) { lbl[i] = m; changed = 1; }
      }
    }
    for (int j = 0; j < 8; ++j) {                // pointer jumping: l[i] <- l[l[i]]
#pragma unroll 2
      for (int k = 0; k < PPT; ++k) {
        int i = t + (k << 10);
        int v = lbl[i];
        if (v < LSENT) {
          int w = lbl[v];
          if (w < v) { lbl[i] = w; changed = 1; }
        }
      }
    }
    if (changed) misc[0] = 1;
    __syncthreads();
    if (misc[0] == 0) break;                     // uniform: read after barrier
    __syncthreads();                             // protect reads from next reset
  }

  // ---- Phase 3: foreground count -> background area --------------------------
  int cnt = 0;
#pragma unroll 4
  for (int k = 0; k < PPT; ++k) cnt += (lbl[t + (k << 10)] < LSENT) ? 1 : 0;
  for (int off = 16; off; off >>= 1) cnt += __shfl_down(cnt, off, 32);
  __syncthreads();
  if (lane == 0) uni[wid] = cnt;
  __syncthreads();
  if (wid == 0) {
    int v = uni[lane];                            // exactly 32 waves
    for (int off = 16; off; off >>= 1) v += __shfl_down(v, off, 32);
    if (lane == 0) misc[1] = v;
  }
  __syncthreads();
  const int fgTotal = misc[1];

  // ---- Phase 4: per-label areas via 5 LDS-histogram passes, track top-2 -------
  Top2 loc; loc.a1 = -1; loc.l1 = 0x7fffffff; loc.a2 = -1; loc.l2 = 0x7fffffff;
  for (int p = 0; p < NPASS; ++p) {
    const int base = p * CH;
    for (int j = t; j < CH; j += NTH) uni[j] = 0;
    __syncthreads();
#pragma unroll 4
    for (int k = 0; k < PPT; ++k) {
      int v = lbl[t + (k << 10)];
      unsigned d = (unsigned)(v - base);
      if (v < MHW && d < (unsigned)CH) atomicAdd(&uni[d], 1);
    }
    __syncthreads();
    for (int j = t; j < CH; j += NTH) {           // zero-area labels included
      int labn = base + j;
      if (labn < MHW) t2_push(loc, uni[j], labn);
    }
    __syncthreads();
  }
  if (t == 0) t2_push(loc, MHW - fgTotal, MHW);   // background = stats row HW
  for (int off = 16; off; off >>= 1) {
    int a1 = __shfl_down(loc.a1, off, 32), l1 = __shfl_down(loc.l1, off, 32);
    int a2 = __shfl_down(loc.a2, off, 32), l2 = __shfl_down(loc.l2, off, 32);
    t2_push(loc, a1, l1); t2_push(loc, a2, l2);
  }
  if (lane == 0) {
    uni[wid * 4 + 0] = loc.a1; uni[wid * 4 + 1] = loc.l1;
    uni[wid * 4 + 2] = loc.a2; uni[wid * 4 + 3] = loc.l2;
  }
  __syncthreads();
  if (wid == 0) {
    Top2 c;
    c.a1 = uni[lane * 4 + 0]; c.l1 = uni[lane * 4 + 1];
    c.a2 = uni[lane * 4 + 2]; c.l2 = uni[lane * 4 + 3];
    for (int off = 16; off; off >>= 1) {
      int a1 = __shfl_down(c.a1, off, 32), l1 = __shfl_down(c.l1, off, 32);
      int a2 = __shfl_down(c.a2, off, 32), l2 = __shfl_down(c.l2, off, 32);
      t2_push(c, a1, l1); t2_push(c, a2, l2);
    }
    if (lane == 0) misc[2] = c.l2;                // second largest overall
  }
  __syncthreads();
  const int lab2 = misc[2];

  // ---- Phase 5: bbox of selected component (works for bg: lab2 == LSENT) ------
  int rmin = MH, rmax = -1, cmin = MW, cmax = -1;
#pragma unroll 4
  for (int k = 0; k < PPT; ++k) {
    int i = t + (k << 10);
    if (lbl[i] == lab2) {
      int r = i >> 8, c = i & 255;
      rmin = min(rmin, r); rmax = max(rmax, r);
      cmin = min(cmin, c); cmax = max(cmax, c);
    }
  }
  for (int off = 16; off; off >>= 1) {
    rmin = min(rmin, __shfl_down(rmin, off, 32));
    rmax = max(rmax, __shfl_down(rmax, off, 32));
    cmin = min(cmin, __shfl_down(cmin, off, 32));
    cmax = max(cmax, __shfl_down(cmax, off, 32));
  }
  if (lane == 0) {
    uni[wid * 4 + 0] = rmin; uni[wid * 4 + 1] = rmax;
    uni[wid * 4 + 2] = cmin; uni[wid * 4 + 3] = cmax;
  }
  __syncthreads();
  if (wid == 0) {
    int a = uni[lane * 4 + 0], b = uni[lane * 4 + 1];
    int c = uni[lane * 4 + 2], d = uni[lane * 4 + 3];
    for (int off = 16; off; off >>= 1) {
      a = min(a, __shfl_down(a, off, 32));
      b = max(b, __shfl_down(b, off, 32));
      c = min(c, __shfl_down(c, off, 32));
      d = max(d, __shfl_down(d, off, 32));
    }
    if (lane == 0) { misc[3] = a; misc[4] = b + 1; misc[5] = c; misc[6] = d + 1; }
  }
  __syncthreads();

  // ---- Phase 6: out = where(box & any(fg), mask, 0) ---------------------------
  const int valid = (fgTotal > 0);
  const int h0 = misc[3], h1 = misc[4], w0 = misc[5], w1 = misc[6];
#pragma unroll 4
  for (int k = 0; k < PPT; ++k) {
    int i = t + (k << 10);
    int r = i >> 8, c = i & 255;
    bool inbox = valid && (r >= h0) && (r < h1) && (c >= w0) && (c < w1);
    dst[i] = inbox ? src[i] : 0.0f;
  }
}

extern "C" void kernel_launch(void* const* d_in, const int* in_sizes, int n_in,
                              void* d_out, int out_size, void* d_ws, size_t ws_size,
                              hipStream_t stream) {
  (void)n_in; (void)out_size; (void)d_ws; (void)ws_size;
  const float* masks = (const float*)d_in[0];
  float* out = (float*)d_out;
  const int nMasks = in_sizes[0] / MHW;           // 8*16 = 128 masks
  (void)hipFuncSetAttribute((const void*)ccl_second_largest_kernel,
                            hipFuncAttributeMaxDynamicSharedMemorySize, SMEM_BYTES);
  hipLaunchKernelGGL(ccl_second_largest_kernel, dim3(nMasks), dim3(NTH),
                     SMEM_BYTES, stream, masks, out);
}